// DynamicSSM_BlockDiagonal_34033320853880
// MI455X (gfx1250) — compile-verified
//
#include <hip/hip_runtime.h>
#include <math.h>

// CDNA5 / gfx1250. wave32. Uses V_WMMA_F32_16X16X4_F32 via builtin.
typedef __attribute__((ext_vector_type(2))) float v2f;
typedef __attribute__((ext_vector_type(8))) float v8f;

#define BATCH  16
#define NFREQ  2048
#define DMODEL 512
#define BS     32
#define NB     16
#define NK     12   // Neumann terms: spectral radius ~0.16 -> 0.16^12 ~ 1e-10

// P layout per (n,k): [re-plane 1024 | im-plane 1024] floats,
// within a plane: element (row i, col j) at ((i>>1)*32 + j)*2 + (i&1)
// -> the two rows of a WMMA B-fragment VGPR pair are 8-byte contiguous.
__device__ __forceinline__ int pidx(int i, int j) {
  return ((i >> 1) * 32 + j) * 2 + (i & 1);
}

// dst = a @ b, 32x32 complex, all interleaved (re,im) row-major in LDS.
__device__ __forceinline__ void cmatmul_lds(float* __restrict__ dst,
                                            const float* __restrict__ a,
                                            const float* __restrict__ b,
                                            int tid) {
  #pragma unroll
  for (int q = 0; q < 4; ++q) {
    int o = q * 256 + tid;
    int i = o >> 5, j = o & 31;
    float accr = 0.f, acci = 0.f;
    for (int t = 0; t < 32; ++t) {
      float ar = a[(i * 32 + t) * 2 + 0];
      float ai = a[(i * 32 + t) * 2 + 1];
      float br = b[(t * 32 + j) * 2 + 0];
      float bi = b[(t * 32 + j) * 2 + 1];
      accr = fmaf(ar, br, accr); accr = fmaf(-ai, bi, accr);
      acci = fmaf(ar, bi, acci); acci = fmaf(ai, br, acci);
    }
    dst[(i * 32 + j) * 2 + 0] = accr;
    dst[(i * 32 + j) * 2 + 1] = acci;
  }
}

__device__ __forceinline__ void storeP(float* __restrict__ Pk,
                                       const float* __restrict__ m, int tid) {
  for (int o = tid; o < 1024; o += 256) {
    int i = o >> 5, j = o & 31;
    int p = pidx(i, j);
    Pk[p]        = m[o * 2 + 0];
    Pk[1024 + p] = m[o * 2 + 1];
  }
}

// Kernel 1: per block n, compute P_0 = C@B, P_k = A@P_{k-1}. Tiny (16 WGs).
__global__ __launch_bounds__(256) void ssm_precompute(
    const float* __restrict__ Ag, const float* __restrict__ Bg,
    const float* __restrict__ Cg, float* __restrict__ P) {
  __shared__ float bufA[2048], buf0[2048], buf1[2048];
  int n = blockIdx.x, tid = threadIdx.x;
  const float* An = Ag + n * 2048;
  const float* Bn = Bg + n * 2048;
  const float* Cn = Cg + n * 2048;
  for (int t = tid; t < 2048; t += 256) { bufA[t] = Cn[t]; buf0[t] = Bn[t]; }
  __syncthreads();
  cmatmul_lds(buf1, bufA, buf0, tid);           // buf1 = C @ B
  __syncthreads();
  for (int t = tid; t < 2048; t += 256) bufA[t] = An[t];   // bufA = A
  float* Pn = P + (size_t)n * NK * 2048;
  storeP(Pn, buf1, tid);                        // P_0
  __syncthreads();
  float* cur = buf1; float* nxt = buf0;
  for (int k = 1; k < NK; ++k) {
    cmatmul_lds(nxt, bufA, cur, tid);           // P_k = A @ P_{k-1}
    __syncthreads();
    storeP(Pn + (size_t)k * 2048, nxt, tid);
    float* t = cur; cur = nxt; nxt = t;
    __syncthreads();
  }
}

// Kernel 2: grid (NFREQ/8, NB), 256 threads (8 waves); wave w handles
// frequency f = blockIdx.x*8 + w. y(16x32) += Sum_k (x * e^{-jkw}) @ P_k
// via f32 WMMA with accumulation chained through the C operand.
__global__ __launch_bounds__(256) void ssm_apply(
    const float* __restrict__ xg, const float* __restrict__ P,
    float* __restrict__ yg) {
  __shared__ float sP[NK * 2048];               // 96 KB: all P_k for block n
  int tid = threadIdx.x;
  int n = blockIdx.y;
  {
    const float4* src = (const float4*)(P + (size_t)n * NK * 2048);
    float4* dst = (float4*)sP;
    #pragma unroll
    for (int t = 0; t < (NK * 2048 / 4) / 256; ++t)
      dst[tid + t * 256] = src[tid + t * 256];
  }
  __syncthreads();

  int lane = tid & 31, wave = tid >> 5;
  int hi = lane >> 4;        // lane half selects K pair (ISA A/B layout)
  int lj = lane & 15;        // M (batch) for A-frag, N (col) for B-frag
  int f = blockIdx.x * 8 + wave;

  // Load x slice as WMMA A-operand fragments: 8 K-chunks of 4.
  v2f xr[8], xi[8], xin[8];
  size_t xbase = (((size_t)lj * NFREQ + (size_t)f) * DMODEL + n * BS) * 2;
  #pragma unroll
  for (int c = 0; c < 8; ++c) {
    float4 v = *(const float4*)(xg + xbase + (size_t)(4 * c + 2 * hi) * 2);
    v2f r, im, ni;
    r.x = v.x;  r.y = v.z;
    im.x = v.y; im.y = v.w;
    ni.x = -v.y; ni.y = -v.w;
    xr[c] = r; xi[c] = im; xin[c] = ni;        // xin = -xi (for Re-part)
  }

  v8f yr0 = {0,0,0,0,0,0,0,0};
  v8f yr1 = yr0, yi0 = yr0, yi1 = yr0;

  // Per-term rotation by e^{-j*2*pi*f/NFREQ}
  float ang = 6.283185307179586f * (float)f / (float)NFREQ;
  float cw = cosf(ang), sw = sinf(ang);

  for (int k = 0; k < NK; ++k) {
    const float* pr  = sP + k * 2048;
    const float* pim = pr + 1024;
    #pragma unroll
    for (int c = 0; c < 8; ++c) {
      int ro = (2 * c + hi) * 64;              // row-pair base in plane
      v2f pr0 = *(const v2f*)(pr  + ro +      lj * 2);
      v2f pr1 = *(const v2f*)(pr  + ro + 32 + lj * 2);
      v2f pi0 = *(const v2f*)(pim + ro +      lj * 2);
      v2f pi1 = *(const v2f*)(pim + ro + 32 + lj * 2);
      // yr += xr@Pr - xi@Pi ; yi += xr@Pi + xi@Pr   (per 16-col tile)
      yr0 = __builtin_amdgcn_wmma_f32_16x16x4_f32(false, xr[c],  false, pr0, (short)0, yr0, false, false);
      yr0 = __builtin_amdgcn_wmma_f32_16x16x4_f32(false, xin[c], false, pi0, (short)0, yr0, false, false);
      yr1 = __builtin_amdgcn_wmma_f32_16x16x4_f32(false, xr[c],  false, pr1, (short)0, yr1, false, false);
      yr1 = __builtin_amdgcn_wmma_f32_16x16x4_f32(false, xin[c], false, pi1, (short)0, yr1, false, false);
      yi0 = __builtin_amdgcn_wmma_f32_16x16x4_f32(false, xr[c],  false, pi0, (short)0, yi0, false, false);
      yi0 = __builtin_amdgcn_wmma_f32_16x16x4_f32(false, xi[c],  false, pr0, (short)0, yi0, false, false);
      yi1 = __builtin_amdgcn_wmma_f32_16x16x4_f32(false, xr[c],  false, pi1, (short)0, yi1, false, false);
      yi1 = __builtin_amdgcn_wmma_f32_16x16x4_f32(false, xi[c],  false, pr1, (short)0, yi1, false, false);
    }
    if (k != NK - 1) {
      // x <- x * (cos - j sin): xr' = c*xr + s*xi ; xi' = c*xi - s*xr
      #pragma unroll
      for (int c = 0; c < 8; ++c) {
        v2f r = xr[c], im = xi[c], nr, ni, nn;
        nr.x = fmaf(cw, r.x,  sw * im.x);
        nr.y = fmaf(cw, r.y,  sw * im.y);
        ni.x = fmaf(cw, im.x, -sw * r.x);
        ni.y = fmaf(cw, im.y, -sw * r.y);
        nn.x = -ni.x; nn.y = -ni.y;
        xr[c] = nr; xi[c] = ni; xin[c] = nn;
      }
    }
  }

  // Store D tiles: VGPR r -> rows (r, r+8) per lane-half; re/im interleave.
  #pragma unroll
  for (int r = 0; r < 8; ++r) {
    int b = r + 8 * hi;
    size_t base = (((size_t)b * NFREQ + (size_t)f) * DMODEL + n * BS) * 2;
    float2 o0; o0.x = yr0[r]; o0.y = yi0[r];
    float2 o1; o1.x = yr1[r]; o1.y = yi1[r];
    *(float2*)(yg + base + (size_t)lj * 2)        = o0;
    *(float2*)(yg + base + (size_t)(lj + 16) * 2) = o1;
  }
}

extern "C" void kernel_launch(void* const* d_in, const int* in_sizes, int n_in,
                              void* d_out, int out_size, void* d_ws, size_t ws_size,
                              hipStream_t stream) {
  const float* x  = (const float*)d_in[0];   // complex64 -> interleaved f32
  const float* A  = (const float*)d_in[1];
  const float* Bm = (const float*)d_in[2];
  const float* C  = (const float*)d_in[3];
  float* Pws = (float*)d_ws;                 // needs NB*NK*2048*4 = 1.5 MB
  (void)in_sizes; (void)n_in; (void)out_size; (void)ws_size;

  ssm_precompute<<<NB, 256, 0, stream>>>(A, Bm, C, Pws);
  ssm_apply<<<dim3(NFREQ / 8, NB), 256, 0, stream>>>(x, Pws, (float*)d_out);
}